// FocalRegionLoss_35673998361264
// MI455X (gfx1250) — compile-verified
//
#include <hip/hip_runtime.h>

#define HW   (512 * 512)
#define BSZ  8
#define CSZ  3
#define RNUM 64
#define BETA_F 1.0f
#define PIX_PER_BLOCK 2048   // 256 threads x 2 iters x 4 pixels (float4)
#define CHUNKS_PER_IMG (HW / PIX_PER_BLOCK)   // 128

typedef __attribute__((ext_vector_type(2))) float v2f;
typedef __attribute__((ext_vector_type(8))) float v8f;

// Wave-level sum of one float per lane via V_WMMA_F32_16X16X4_F32.
// A = {partial, 0} per lane (16x4), B = ones (4x16):
//   D[m][n] = sum_k A[m][k]  (row sums, replicated across n)
// Lane L (L<16) sums D VGPRs 0..7 = rows 0..7; lane L>=16 gets rows 8..15.
// shfl_xor(16) combines the two halves -> full wave sum in every lane.
// Must be called with EXEC all ones (uniform control flow, full wave).
__device__ __forceinline__ float wave_sum_wmma(float partial) {
    v2f a; a[0] = partial; a[1] = 0.0f;
    v2f b; b[0] = 1.0f;    b[1] = 1.0f;
    v8f c = {};
    c = __builtin_amdgcn_wmma_f32_16x16x4_f32(
        /*neg_a=*/false, a, /*neg_b=*/false, b,
        /*c_mod=*/(short)0, c, /*reuse_a=*/false, /*reuse_b=*/false);
    float s = c[0] + c[1] + c[2] + c[3] + c[4] + c[5] + c[6] + c[7];
    s += __shfl_xor(s, 16);
    return s;
}

__global__ void __launch_bounds__(256) frl_init(float* S, int* N, float* T) {
    int i = blockIdx.x * blockDim.x + threadIdx.x;
    if (i < BSZ * RNUM) { S[i] = 0.0f; N[i] = 0; }
    if (i == 0) T[0] = 0.0f;
}

__global__ void __launch_bounds__(256) frl_main(const float* __restrict__ inp,
                                                const float* __restrict__ tgt,
                                                const int*   __restrict__ mask,
                                                float* __restrict__ S,
                                                int*   __restrict__ N,
                                                float* __restrict__ T) {
    __shared__ float sbins[RNUM];
    __shared__ int   scnt[RNUM];
    __shared__ float stot;
    const int t = threadIdx.x;
    if (t < RNUM) { sbins[t] = 0.0f; scnt[t] = 0; }
    if (t == 0) stot = 0.0f;
    __syncthreads();

    const int blk = blockIdx.x;
    const int b   = blk >> 7;                       // CHUNKS_PER_IMG == 128
    const int hw0 = (blk & (CHUNKS_PER_IMG - 1)) * PIX_PER_BLOCK;

    const float* ip = inp  + (size_t)b * CSZ * HW;
    const float* tp = tgt  + (size_t)b * CSZ * HW;
    const int*   mp = mask + (size_t)b * HW;

    float tot = 0.0f;
#pragma unroll
    for (int k = 0; k < 2; ++k) {
        const int hw = hw0 + (k * 256 + t) * 4;
        const float4 i0 = *(const float4*)(ip + hw);
        const float4 i1 = *(const float4*)(ip + HW + hw);
        const float4 i2 = *(const float4*)(ip + 2 * HW + hw);
        const float4 t0 = *(const float4*)(tp + hw);
        const float4 t1 = *(const float4*)(tp + HW + hw);
        const float4 t2 = *(const float4*)(tp + 2 * HW + hw);
        const int4   m  = *(const int4*)(mp + hw);

        float cs0 = fabsf(i0.x - t0.x) + fabsf(i1.x - t1.x) + fabsf(i2.x - t2.x);
        float cs1 = fabsf(i0.y - t0.y) + fabsf(i1.y - t1.y) + fabsf(i2.y - t2.y);
        float cs2 = fabsf(i0.z - t0.z) + fabsf(i1.z - t1.z) + fabsf(i2.z - t2.z);
        float cs3 = fabsf(i0.w - t0.w) + fabsf(i1.w - t1.w) + fabsf(i2.w - t2.w);

        tot += cs0 + cs1 + cs2 + cs3;
        atomicAdd(&sbins[m.x], cs0); atomicAdd(&scnt[m.x], 1);
        atomicAdd(&sbins[m.y], cs1); atomicAdd(&scnt[m.y], 1);
        atomicAdd(&sbins[m.z], cs2); atomicAdd(&scnt[m.z], 1);
        atomicAdd(&sbins[m.w], cs3); atomicAdd(&scnt[m.w], 1);
    }

    // uniform control flow here: EXEC all ones -> WMMA wave reduction
    float wsum = wave_sum_wmma(tot);
    if ((t & 31) == 0) atomicAdd(&stot, wsum);
    __syncthreads();

    if (t < RNUM) {
        atomicAdd(&S[b * RNUM + t], sbins[t]);
        atomicAdd(&N[b * RNUM + t], scnt[t]);
    }
    if (t == 0) atomicAdd(T, stot);
}

__global__ void __launch_bounds__(512) frl_final(const float* __restrict__ S,
                                                 const int*   __restrict__ N,
                                                 const float* __restrict__ T,
                                                 float* __restrict__ out) {
    __shared__ float red[BSZ * RNUM];
    const int t = threadIdx.x;           // 512 threads
    const float s = S[t];
    const float mean = s / fmaxf((float)N[t] * (float)CSZ, 1.0f);

    // max over all region means (empty regions contribute 0, matching the
    // reference: their mean is 0/1 = 0 and weights are nonnegative)
    red[t] = mean;
    __syncthreads();
    for (int off = 256; off > 0; off >>= 1) {
        if (t < off) red[t] = fmaxf(red[t], red[t + off]);
        __syncthreads();
    }
    const float maxw = red[0];
    __syncthreads();

    float w = (maxw > 0.0f) ? (mean / maxw) : mean;
    w = fminf(fmaxf(w, 0.0f), 1.0f);

    // sum_{b,r} w[b,r] * S[b,r]  ==  sum over all pixels of chan_sum * w
    red[t] = w * s;
    __syncthreads();
    for (int off = 256; off > 0; off >>= 1) {
        if (t < off) red[t] += red[t + off];
        __syncthreads();
    }
    if (t == 0) {
        const float total = (float)((size_t)BSZ * CSZ * HW);
        out[0] = (T[0] + BETA_F * red[0]) / total;
    }
}

extern "C" void kernel_launch(void* const* d_in, const int* in_sizes, int n_in,
                              void* d_out, int out_size, void* d_ws, size_t ws_size,
                              hipStream_t stream) {
    (void)in_sizes; (void)n_in; (void)out_size; (void)ws_size;
    const float* inp  = (const float*)d_in[0];
    const float* tgt  = (const float*)d_in[1];
    const int*   mask = (const int*)d_in[2];
    float* out = (float*)d_out;

    // scratch layout: S[512] floats | N[512] ints | T[1] float
    float* S = (float*)d_ws;
    int*   N = (int*)(S + BSZ * RNUM);
    float* T = (float*)(N + BSZ * RNUM);

    frl_init<<<2, 256, 0, stream>>>(S, N, T);
    frl_main<<<BSZ * CHUNKS_PER_IMG, 256, 0, stream>>>(inp, tgt, mask, S, N, T);
    frl_final<<<1, BSZ * RNUM, 0, stream>>>(S, N, T, out);
}